// ConvC2_44581760532829
// MI455X (gfx1250) — compile-verified
//
#include <hip/hip_runtime.h>
#include <hip/hip_bf16.h>

typedef __attribute__((ext_vector_type(2))) float v2f;
typedef __attribute__((ext_vector_type(8))) float v8f;

#define H_IN   512
#define W_IN   512
#define P_OUT  508
#define Q_OUT  508
#define KREAL  150   // 6 channels * 25 taps
#define KPAD   152   // padded to multiple of 4
#define NCHUNK 38    // KPAD / 4
#define TP     8     // output rows per workgroup
#define TQ     128   // output cols per workgroup (8 waves * 16 px)
#define XROWS  12    // TP + R - 1
#define XCOLS  132   // TQ + S - 1

// LeNet C3 connectivity: bit m of MASKBITS[c] == 1 iff input channel c feeds output m.
__device__ __constant__ unsigned int MASKBITS[6] = {
    0xDE71u, 0xBCE3u, 0xE9C7u, 0xD3CEu, 0xB79Cu, 0xEF38u};

// Compile-time LDS offset for flat K index -> (c, r, s) within the input tile.
// xlds layout: [(c*XROWS + row)*XCOLS + col]; the (pl, qloc) part is the runtime base.
__device__ __forceinline__ constexpr int boff(int k) {
    if (k >= KREAL) return 0;           // tail pad: A is zero there, any valid addr ok
    const int c  = k / 25;
    const int rs = k % 25;
    const int r  = rs / 5;
    const int s  = rs % 5;
    return c * (XROWS * XCOLS) + r * XCOLS + s;
}

__global__ __launch_bounds__(256) void conv_c3_wmma(
    const float* __restrict__ x,     // [64, 6, 512, 512]
    const float* __restrict__ Wt,    // [16, 6, 5, 5]
    const float* __restrict__ bias,  // [16]
    float* __restrict__ out)         // [64, 16, 508, 508]
{
    __shared__ float xlds[6 * XROWS * XCOLS];  // 9504 floats = 38.0 KB
    __shared__ float wlds[KPAD * 16];          // 2432 floats =  9.5 KB, [K][16] masked+padded
    __shared__ float blds[16];

    const int tid = (int)threadIdx.x;
    const int n   = (int)blockIdx.z;
    const int p0  = (int)blockIdx.y * TP;
    const int q0  = (int)blockIdx.x * TQ;

    // ---- Stage masked, K-padded weights into LDS as [K][M=16] ----
    for (int idx = tid; idx < KPAD * 16; idx += 256) {
        const int k = idx >> 4;
        const int m = idx & 15;
        float v = 0.0f;
        if (k < KREAL) {
            const int c = k / 25;                            // k = c*25 + r*5 + s
            v = Wt[m * KREAL + k] * (float)((MASKBITS[c] >> m) & 1u);
        }
        wlds[idx] = v;
    }

    // ---- Stage input tile: 6ch x 12rows x 132cols, edge-clamped ----
    const size_t nbase = (size_t)n * 6 * H_IN * W_IN;
    for (int idx = tid; idx < 6 * XROWS * XCOLS; idx += 256) {
        const int col = idx % XCOLS;
        const int t   = idx / XCOLS;
        const int row = t % XROWS;
        const int c   = t / XROWS;
        int gr = p0 + row; if (gr > H_IN - 1) gr = H_IN - 1;
        int gc = q0 + col; if (gc > W_IN - 1) gc = W_IN - 1;
        xlds[idx] = x[nbase + (size_t)c * (H_IN * W_IN) + (size_t)gr * W_IN + gc];
    }
    if (tid < 16) blds[tid] = bias[tid];
    __syncthreads();

    const int  lane = tid & 31;
    const int  wid  = tid >> 5;
    const int  m    = lane & 15;       // A: channel row; B: pixel column
    const bool hi   = lane >= 16;      // lane half selects K pair {0,1} vs {2,3}

    // ---- Hoist all A fragments (weights) into registers: reused for 8 rows ----
    float a0[NCHUNK], a1[NCHUNK];
#pragma unroll
    for (int u = 0; u < NCHUNK; ++u) {
        const int kk = 4 * u + (hi ? 2 : 0);
        a0[u] = wlds[kk * 16 + m];
        a1[u] = wlds[kk * 16 + 16 + m];
    }

    const size_t obase_n = (size_t)n * 16 * (P_OUT * Q_OUT);
    const int    qloc    = wid * 16 + m;
    const int    q       = q0 + qloc;
    const bool   qok     = q < Q_OUT;

    // Fully unrolled over the 8 output rows: pl*XCOLS folds into the 16-bit
    // ds_load instruction offset, so the steady state is ds_load + v_wmma only.
#pragma unroll
    for (int pl = 0; pl < TP; ++pl) {
        v8f acc = {0.f, 0.f, 0.f, 0.f, 0.f, 0.f, 0.f, 0.f};
#pragma unroll
        for (int u = 0; u < NCHUNK; ++u) {
            v2f av; av.x = a0[u]; av.y = a1[u];
            v2f bv;
            bv.x = xlds[pl * XCOLS + qloc + (hi ? boff(4 * u + 2) : boff(4 * u))];
            bv.y = xlds[pl * XCOLS + qloc + (hi ? boff(4 * u + 3) : boff(4 * u + 1))];
            // D = A(16x4 f32) * B(4x16 f32) + C(16x16 f32)
            acc = __builtin_amdgcn_wmma_f32_16x16x4_f32(
                false, av, false, bv, (short)0, acc, false, false);
        }

        const int p = p0 + pl;
        if (p < P_OUT && qok) {
            const size_t ob = obase_n + (size_t)p * Q_OUT + q;
#pragma unroll
            for (int i = 0; i < 8; ++i) {
                const int ch = i + (hi ? 8 : 0);   // C/D layout: VGPR i, lane-half -> M
                out[ob + (size_t)ch * (P_OUT * Q_OUT)] = acc[i] + blds[ch];
            }
        }
    }
}

extern "C" void kernel_launch(void* const* d_in, const int* in_sizes, int n_in,
                              void* d_out, int out_size, void* d_ws, size_t ws_size,
                              hipStream_t stream) {
    const float* x  = (const float*)d_in[0];
    const float* Wt = (const float*)d_in[1];
    const float* b  = (const float*)d_in[2];
    float* out      = (float*)d_out;

    dim3 grid((Q_OUT + TQ - 1) / TQ,   // 4
              (P_OUT + TP - 1) / TP,   // 64
              64);                     // batch
    dim3 block(256);
    conv_c3_wmma<<<grid, block, 0, stream>>>(x, Wt, b, out);
}